// PixelRNN_45715631899296
// MI455X (gfx1250) — compile-verified
//
#include <hip/hip_runtime.h>

// ---------------------------------------------------------------------------
// PixelRNN on CDNA5: 256-step LSTM recurrence, batch-parallel waves,
// f32 WMMA 16x16x4 for gate GEMMs AND the output head.
// Weights pre-packed into WMMA B-lane layout; compiler hoists them into the
// wave32 extended register file (>256 VGPRs via s_set_vgpr_msb), so the
// steady-state loop is LDS A-loads + v_wmma only.
// ---------------------------------------------------------------------------

typedef __attribute__((ext_vector_type(2))) float v2f;
typedef __attribute__((ext_vector_type(8))) float v8f;

#define HID        64
#define KTOT       88          // 3*RS (=24) feed + 64 hidden
#define KT         22          // KTOT / 4 (K per f32 WMMA)
#define HKT        16          // HID / 4 (head K-tiles)
#define ACT_STRIDE 90          // 88 + 2 pad, even (ds_load_b64 alignment)
#define ROW_STRIDE 72          // 3 channels * 24 columns
#define WAVE_LDS   (16 * ACT_STRIDE + 16 * ROW_STRIDE)   // 2592 floats / wave

#define NPACK_B    (16 * KT * 64)     // gate weights: 22528 floats
#define NPACK_HEAD (HKT * 64)         // head weights: 1024 floats

__device__ __forceinline__ float sigf(float x) {
  return 1.0f / (1.0f + __expf(-x));
}

__device__ __forceinline__ float tanh_fast(float x) {
#if __has_builtin(__builtin_amdgcn_tanhf)
  return __builtin_amdgcn_tanhf(x);        // v_tanh_f32 (CDNA5 trans op)
#else
  float e = __expf(-2.0f * x);
  return (1.0f - e) / (1.0f + e);
#endif
}

// ---------------------------------------------------------------------------
// Pack weights into WMMA B-operand lane layout.
//  packB  : tile t = ntile*22 + ktile, entry lane*2+v holds
//           B[k = ktile*4 + (lane>=16?2:0) + v][n = ntile*16 + (lane&15)]
//           from Wcomb[n][k] = k<24 ? W_ih[n][k] : W_hh[n][k-24]
//  bsum   : b_ih + b_hh
//  packHd : head panel, 16 K-tiles of Wl^T padded 64x3 -> 64x16 (cols>=3 zero)
// ---------------------------------------------------------------------------
__global__ void pixelrnn_pack(const float* __restrict__ W_ih,
                              const float* __restrict__ W_hh,
                              const float* __restrict__ b_ih,
                              const float* __restrict__ b_hh,
                              const float* __restrict__ Wl,
                              float* __restrict__ packB,
                              float* __restrict__ bsum,
                              float* __restrict__ packHd) {
  int tid = blockIdx.x * blockDim.x + threadIdx.x;
  if (tid < NPACK_B) {
    int t     = tid >> 6;
    int e     = tid & 63;
    int lane  = e >> 1;
    int v     = e & 1;
    int ntile = t / KT;
    int kt    = t - ntile * KT;
    int n     = ntile * 16 + (lane & 15);
    int k     = kt * 4 + ((lane >> 4) << 1) + v;
    packB[tid] = (k < 24) ? W_ih[n * 24 + k] : W_hh[n * 64 + (k - 24)];
  } else if (tid < NPACK_B + 256) {
    int j = tid - NPACK_B;
    bsum[j] = b_ih[j] + b_hh[j];
  } else if (tid < NPACK_B + 256 + NPACK_HEAD) {
    int idx  = tid - (NPACK_B + 256);
    int kt   = idx >> 6;
    int e    = idx & 63;
    int lane = e >> 1;
    int v    = e & 1;
    int n    = lane & 15;                       // channel (pad >= 3)
    int k    = kt * 4 + ((lane >> 4) << 1) + v; // 0..63
    packHd[idx] = (n < 3) ? Wl[n * HID + k] : 0.0f;
  }
}

// ---------------------------------------------------------------------------
// Main kernel: each wave owns 16 batch elements and runs the full 16x16
// recurrence. 4 independent waves per block; no block-level sync needed.
// ---------------------------------------------------------------------------
__global__ __launch_bounds__(128) void pixelrnn_kernel(
    const float* __restrict__ x,       // [B,3,24,24]
    const float* __restrict__ bl,      // [3]
    const float* __restrict__ packB,   // [16*22*64]
    const float* __restrict__ bsum,    // [256]
    const float* __restrict__ packHd,  // [16*64]
    float* __restrict__ out)           // [B,3,16,16]
{
  __shared__ float smem[4 * WAVE_LDS];
  const int lane = threadIdx.x & 31;
  const int wave = threadIdx.x >> 5;
  const int tile = blockIdx.x * 4 + wave;
  const int b0   = tile * 16;

  float* act    = smem + wave * WAVE_LDS;     // [16][ACT_STRIDE]: feed(0..23)+hx(24..87)
  float* rowbuf = act + 16 * ACT_STRIDE;      // [16][3][24]

  const int mlo     = lane & 15;              // A-row / D-column index
  const int khalf   = (lane >> 4) << 1;       // K offset for A/B lane-half
  const int mhi_off = (lane >> 4) << 3;       // D-layout: lanes 16-31 hold M+8

  // hx = 0
  for (int j = lane; j < 16 * HID; j += 32)
    act[(j >> 6) * ACT_STRIDE + 24 + (j & 63)] = 0.0f;

  // cx = 0 (registers, WMMA D layout)
  v8f cx[4];
#pragma unroll
  for (int t = 0; t < 4; ++t)
#pragma unroll
    for (int r = 0; r < 8; ++r) cx[t][r] = 0.0f;

  // per-lane gate bias (constant over steps): n = g*64 + ht*16 + mlo
  float bias[4][4];
#pragma unroll
  for (int ht = 0; ht < 4; ++ht)
#pragma unroll
    for (int g = 0; g < 4; ++g) bias[ht][g] = bsum[g * 64 + ht * 16 + mlo];

  // head bias per lane (D-column = channel; pad columns get 0)
  const float blv = (mlo < 3) ? bl[mlo] : 0.0f;

  for (int row = 0; row < 16; ++row) {
    // Load image row (row+8) of x into the row buffer.
    for (int j = lane; j < 16 * ROW_STRIDE; j += 32) {
      int m   = j / ROW_STRIDE;
      int rem = j - m * ROW_STRIDE;           // ch*24 + w
      rowbuf[m * ROW_STRIDE + rem] =
          x[(size_t)(b0 + m) * 1728 + (rem / 24) * 576 + (row + 8) * 24 + (rem % 24)];
    }

    for (int col = 8; col < 24; ++col) {
      // Rebuild feed window: act[m][k] = rowbuf[m][k/8][col-8 + k%8]
      for (int j = lane; j < 16 * 24; j += 32) {
        int m = j / 24;
        int k = j - m * 24;
        act[m * ACT_STRIDE + k] =
            rowbuf[m * ROW_STRIDE + (k >> 3) * 24 + (col - 8) + (k & 7)];
      }

      // gates[16,256] = act[16,88] @ Wcomb^T + bias; per hidden-tile the
      // i/f/g/o panels form 4 independent WMMA accumulation chains.
#pragma unroll
      for (int ht = 0; ht < 4; ++ht) {
        v8f acc[4];
#pragma unroll
        for (int g = 0; g < 4; ++g) {
          float bv = bias[ht][g];
#pragma unroll
          for (int r = 0; r < 8; ++r) acc[g][r] = bv;
        }

        for (int kt = 0; kt < KT; ++kt) {
          v2f a = *(const v2f*)&act[mlo * ACT_STRIDE + kt * 4 + khalf];
#pragma unroll
          for (int g = 0; g < 4; ++g) {
            const int ntile = g * 4 + ht;
            v2f b = *(const v2f*)(packB + (size_t)(ntile * KT + kt) * 64 + lane * 2);
            acc[g] = __builtin_amdgcn_wmma_f32_16x16x4_f32(
                false, a, false, b, (short)0, acc[g], false, false);
          }
        }

        // Pointwise LSTM: lane holds (m = r+mhi_off, n = mlo)
#pragma unroll
        for (int r = 0; r < 8; ++r) {
          float ig = sigf(acc[0][r]);
          float fg = sigf(acc[1][r]);
          float gg = tanh_fast(acc[2][r]);
          float og = sigf(acc[3][r]);
          float c  = fg * cx[ht][r] + ig * gg;
          cx[ht][r] = c;
          float h = og * tanh_fast(c);
          act[(r + mhi_off) * ACT_STRIDE + 24 + ht * 16 + mlo] = h;  // hx
        }
      }

      // Output head as WMMA: v[16,16pad] = hx[16,64] @ Wl^T(padded) + bl
      {
        v8f hv;
#pragma unroll
        for (int r = 0; r < 8; ++r) hv[r] = blv;
        for (int kt = 0; kt < HKT; ++kt) {
          v2f a = *(const v2f*)&act[mlo * ACT_STRIDE + 24 + kt * 4 + khalf];
          v2f b = *(const v2f*)(packHd + (size_t)kt * 64 + lane * 2);
          hv = __builtin_amdgcn_wmma_f32_16x16x4_f32(
              false, a, false, b, (short)0, hv, false, false);
        }
        // LeakyReLU + scatter (divergence only here, after all WMMAs)
        if (mlo < 3) {
#pragma unroll
          for (int r = 0; r < 8; ++r) {
            int m   = r + mhi_off;
            float v = hv[r];
            v = (v >= 0.0f) ? v : 0.01f * v;
            rowbuf[m * ROW_STRIDE + mlo * 24 + col] = v;   // feedback
            out[(size_t)((b0 + m) * 3 + mlo) * 256 + row * 16 + (col - 8)] = v;
          }
        }
      }
    }
  }
}

// ---------------------------------------------------------------------------
extern "C" void kernel_launch(void* const* d_in, const int* in_sizes, int n_in,
                              void* d_out, int out_size, void* d_ws, size_t ws_size,
                              hipStream_t stream) {
  const float* x    = (const float*)d_in[0];
  const float* W_ih = (const float*)d_in[1];
  const float* W_hh = (const float*)d_in[2];
  const float* b_ih = (const float*)d_in[3];
  const float* b_hh = (const float*)d_in[4];
  const float* Wl   = (const float*)d_in[5];
  const float* bl   = (const float*)d_in[6];
  float* out = (float*)d_out;

  float* packB  = (float*)d_ws;                // 22528 floats
  float* bsum   = packB + NPACK_B;             // 256 floats
  float* packHd = bsum + 256;                  // 1024 floats

  const int B      = in_sizes[0] / (3 * 24 * 24);   // 4096
  const int npack  = NPACK_B + 256 + NPACK_HEAD;
  const int ntiles = B / 16;                        // one wave per 16 batch

  pixelrnn_pack<<<(npack + 255) / 256, 256, 0, stream>>>(
      W_ih, W_hh, b_ih, b_hh, Wl, packB, bsum, packHd);
  pixelrnn_kernel<<<ntiles / 4, 128, 0, stream>>>(
      x, bl, packB, bsum, packHd, out);
}